// Model_29618094473950
// MI455X (gfx1250) — compile-verified
//
#include <hip/hip_runtime.h>
#include <math.h>

// ---------------------------------------------------------------------------
// MI455X (gfx1250) implementation.
//  - Fused WMMA f32 GEMM + LeakyReLU + row L2-normalize (wave32, 16x64 tile/wave)
//  - SpMM via global_atomic_add_f32 scatter (inline asm, guaranteed HW fp atomic)
//  - Modal mixing folded into SpMM scale factors (softmax computed on the fly)
// Workspace layout (floats): Xi[I*64] | Xt[I*64] | M[N*64]   (~64 MB)
// ---------------------------------------------------------------------------

typedef __attribute__((ext_vector_type(2))) float v2f;
typedef __attribute__((ext_vector_type(8))) float v8f;

#define DD 64            // embedding dim
#define LEAK 0.2f
#define MODAL_ADJ_W 0.2f
#define RESIDUAL_W 0.5f

// One wave (32 lanes) computes a full 16-row x 64-col output tile:
//   acc[t] (t=0..3) is the 16x16 f32 WMMA C/D tile for columns [16t,16t+16).
// K-loop uses V_WMMA_F32_16X16X4_F32. Epilogue: LeakyReLU, then per-row
// sum-of-squares via half-wave shfl_xor reduction (lanes 0-15 / 16-31 hold
// disjoint row halves in the C/D layout), then normalized store.
template <int K>
__global__ __launch_bounds__(32) void gemm_lrelu_l2norm(
    const float* __restrict__ X,   // [rows, K] row-major
    const float* __restrict__ W,   // [K, 64]  row-major
    float* __restrict__ Out)       // [rows, 64] normalized
{
    const int row0 = blockIdx.x * 16;
    const int lane = threadIdx.x;
    const int hf   = lane >> 4;    // half-wave: selects K offset (+2) for A/B frags
    const int l16  = lane & 15;    // M index for A frag, N index for B frag

    v8f acc0 = {}, acc1 = {}, acc2 = {}, acc3 = {};

    const float* arow = X + (size_t)(row0 + l16) * K;   // this lane's A row
    __builtin_prefetch(arow, 0, 3);                      // global_prefetch_b8

    for (int k = 0; k < K; k += 4) {
        // A fragment 16x4: lane(l16,hf) holds A[row0+l16][k+2*hf + {0,1}]
        const int ka = k + 2 * hf;
        v2f a;
        a.x = arow[ka];
        a.y = arow[ka + 1];

        // B fragment 4x16 per N-tile: lane holds W[k+2*hf + {0,1}][16t + l16]
        const float* bk = W + (size_t)ka * DD;
        v2f b0, b1, b2, b3;
        b0.x = bk[l16];          b0.y = bk[DD + l16];
        b1.x = bk[16 + l16];     b1.y = bk[DD + 16 + l16];
        b2.x = bk[32 + l16];     b2.y = bk[DD + 32 + l16];
        b3.x = bk[48 + l16];     b3.y = bk[DD + 48 + l16];

        acc0 = __builtin_amdgcn_wmma_f32_16x16x4_f32(false, a, false, b0,
                                                     (short)0, acc0, false, false);
        acc1 = __builtin_amdgcn_wmma_f32_16x16x4_f32(false, a, false, b1,
                                                     (short)0, acc1, false, false);
        acc2 = __builtin_amdgcn_wmma_f32_16x16x4_f32(false, a, false, b2,
                                                     (short)0, acc2, false, false);
        acc3 = __builtin_amdgcn_wmma_f32_16x16x4_f32(false, a, false, b3,
                                                     (short)0, acc3, false, false);
    }

    // Epilogue: LeakyReLU then row-wise L2 normalize, all in registers.
    // C/D layout: vgpr r, lane l -> row = r + 8*(l/16), col = 16t + (l%16).
    float invn[8];
#pragma unroll
    for (int r = 0; r < 8; ++r) {
        float x0 = acc0[r]; x0 = (x0 >= 0.f) ? x0 : LEAK * x0; acc0[r] = x0;
        float x1 = acc1[r]; x1 = (x1 >= 0.f) ? x1 : LEAK * x1; acc1[r] = x1;
        float x2 = acc2[r]; x2 = (x2 >= 0.f) ? x2 : LEAK * x2; acc2[r] = x2;
        float x3 = acc3[r]; x3 = (x3 >= 0.f) ? x3 : LEAK * x3; acc3[r] = x3;
        float s = x0 * x0 + x1 * x1 + x2 * x2 + x3 * x3;
        // Reduce across the 16 lanes of this half-wave (masks 1..8 stay in-half).
        s += __shfl_xor(s, 1, 32);
        s += __shfl_xor(s, 2, 32);
        s += __shfl_xor(s, 4, 32);
        s += __shfl_xor(s, 8, 32);
        invn[r] = 1.0f / fmaxf(sqrtf(s), 1e-12f);
    }
#pragma unroll
    for (int r = 0; r < 8; ++r) {
        float* orow = Out + (size_t)(row0 + r + 8 * hf) * DD + l16;
        orow[0]  = acc0[r] * invn[r];
        orow[16] = acc1[r] * invn[r];
        orow[32] = acc2[r] * invn[r];
        orow[48] = acc3[r] * invn[r];
    }
}

// SpMM scatter: out[rows[e]] += scale * vals[e] * x[cols[e]], x is the virtual
// concat [top(split rows) ; bot]. One wave per edge; lane handles cols 2l,2l+1.
// scale = 1 when widx < 0, else MODAL_ADJ_W * softmax(wvec)[widx].
// Atomic traffic stays in the 192MB L2 (out is only N*64*4 = 38 MB).
__global__ __launch_bounds__(256) void spmm_atomic(
    const int* __restrict__ rows, const int* __restrict__ cols,
    const float* __restrict__ vals, int nnz,
    const float* __restrict__ top, const float* __restrict__ bot, int split,
    const float* __restrict__ wvec, int widx,
    float* __restrict__ out)
{
    const int wave  = (blockIdx.x * blockDim.x + threadIdx.x) >> 5;
    const int lane  = threadIdx.x & 31;

    float scale = 1.0f;
    if (widx >= 0) {
        const float w0 = wvec[0], w1 = wvec[1];
        const float m  = fmaxf(w0, w1);
        const float e0 = __expf(w0 - m), e1 = __expf(w1 - m);
        scale = MODAL_ADJ_W * ((widx == 0 ? e0 : e1) / (e0 + e1));
    }

    if (wave >= nnz) return;
    const int   r = rows[wave];
    const int   c = cols[wave];
    const float v = vals[wave] * scale;

    const float* src = (c < split) ? (top + (size_t)c * DD)
                                   : (bot + (size_t)(c - split) * DD);
    const float x0 = src[2 * lane];
    const float x1 = src[2 * lane + 1];
    const float d0 = v * x0;
    const float d1 = v * x1;

    float* p = out + (size_t)r * DD + 2 * lane;
    // Guaranteed hardware fp32 atomics (no CAS loop), no-return form -> STOREcnt.
    asm volatile("global_atomic_add_f32 %0, %1, off\n\t"
                 "global_atomic_add_f32 %2, %3, off"
                 :
                 : "v"(p), "v"(d0), "v"(p + 1), "v"(d1)
                 : "memory");
}

// out = out + a * x   (out already holds spmm(adj, modal))
__global__ __launch_bounds__(256) void axpy_f4(
    float4* __restrict__ out, const float4* __restrict__ x, float a, int n4)
{
    const int i = blockIdx.x * blockDim.x + threadIdx.x;
    if (i < n4) {
        float4 o = out[i];
        const float4 v = x[i];
        o.x += a * v.x; o.y += a * v.y; o.z += a * v.z; o.w += a * v.w;
        out[i] = o;
    }
}

extern "C" void kernel_launch(void* const* d_in, const int* in_sizes, int n_in,
                              void* d_out, int out_size, void* d_ws, size_t ws_size,
                              hipStream_t stream)
{
    const float* u_embs  = (const float*)d_in[0];
    const float* i_embs  = (const float*)d_in[1];
    const float* img_emb = (const float*)d_in[2];
    const float* txt_emb = (const float*)d_in[3];
    const float* img_mat = (const float*)d_in[4];
    const float* txt_mat = (const float*)d_in[5];
    const float* modal_w = (const float*)d_in[6];
    const int*   adj_r   = (const int*)d_in[7];
    const int*   adj_c   = (const int*)d_in[8];
    const float* adj_v   = (const float*)d_in[9];
    const int*   iadj_r  = (const int*)d_in[10];
    const int*   iadj_c  = (const int*)d_in[11];
    const float* iadj_v  = (const float*)d_in[12];
    const int*   tadj_r  = (const int*)d_in[13];
    const int*   tadj_c  = (const int*)d_in[14];
    const float* tadj_v  = (const float*)d_in[15];

    const int U   = in_sizes[0] / DD;   // 100000
    const int I   = in_sizes[1] / DD;   // 50000
    const int N   = U + I;              // 150000
    const int NNZ = in_sizes[7];        // 2,000,000

    float* Xi  = (float*)d_ws;              // [I,64] normalized image feats
    float* Xt  = Xi + (size_t)I * DD;       // [I,64] normalized text feats
    float* M   = Xt + (size_t)I * DD;       // [N,64] modal accumulator
    float* out = (float*)d_out;             // [N,64] final embeddings

    hipMemsetAsync(M,   0, (size_t)N * DD * sizeof(float), stream);
    hipMemsetAsync(out, 0, (size_t)N * DD * sizeof(float), stream);

    // Fused GEMM + LeakyReLU + L2-normalize (WMMA f32 16x16x4).
    gemm_lrelu_l2norm<1024><<<I / 16, 32, 0, stream>>>(img_emb, img_mat, Xi);
    gemm_lrelu_l2norm<768> <<<I / 16, 32, 0, stream>>>(txt_emb, txt_mat, Xt);

    const int spmm_blocks = (NNZ * 32 + 255) / 256;   // 1 wave per edge

    // M = spmm(adj, [u;i])  (shared by image_aware and text_aware; the
    //     reference's duplicated spmm and the w0+w1==1 softmax fold into x1)
    spmm_atomic<<<spmm_blocks, 256, 0, stream>>>(
        adj_r, adj_c, adj_v, NNZ, u_embs, i_embs, U, nullptr, -1, M);
    // M += 0.2*w0 * spmm(image_adj, [u; norm(img_feats)])
    spmm_atomic<<<spmm_blocks, 256, 0, stream>>>(
        iadj_r, iadj_c, iadj_v, NNZ, u_embs, Xi, U, modal_w, 0, M);
    // M += 0.2*w1 * spmm(text_adj, [u; norm(txt_feats)])
    spmm_atomic<<<spmm_blocks, 256, 0, stream>>>(
        tadj_r, tadj_c, tadj_v, NNZ, u_embs, Xt, U, modal_w, 1, M);

    // out = spmm(adj, M)
    spmm_atomic<<<spmm_blocks, 256, 0, stream>>>(
        adj_r, adj_c, adj_v, NNZ, M, M, 0, nullptr, -1, out);

    // out += (1 + RESIDUAL_W) * M
    const int n4 = N * (DD / 4);
    axpy_f4<<<(n4 + 255) / 256, 256, 0, stream>>>(
        (float4*)out, (const float4*)M, 1.0f + RESIDUAL_W, n4);
}